// ClusterLoss_Regr_88536455840508
// MI455X (gfx1250) — compile-verified
//
#include <hip/hip_runtime.h>
#include <stdint.h>

// ---------------------------------------------------------------------------
// ClusterLoss (regression variant):
//   mask   = |proto_class[p] - label[b]| <= 0.5
//   inv    = (128 - md[b,p]) * mask
//   out    = mean_b(128 - max_p inv)            (top_k, k=1  ==  row max)
//
// Bandwidth-bound streaming reduction (128 MiB read -> 1 scalar; ~0.75 FLOP/B
// => HBM floor ~5.8us at 23.3 TB/s; fits in 192MB L2 for graph replays).
// No contraction anywhere => WMMA inapplicable. CDNA5 path used instead:
// GLOBAL_LOAD_ASYNC_TO_LDS_B128 double-buffered pipeline with the gfx1250
// split counters (s_wait_asynccnt / s_wait_dscnt).
// ---------------------------------------------------------------------------

#define B_ROWS   8192
#define P_COLS   4096
#define TPB      256
#define VEC      4
#define CHUNK    (TPB * VEC)          // 1024 floats = 4 KB per chunk
#define NCHUNK   (P_COLS / CHUNK)     // 4 chunks per row
#define MAX_DIST 128.0f
#define DELTA    0.5f

// Generic pointers into LDS carry the wave-relative LDS byte offset in the
// low 32 bits (aperture base lives in addr[63:32] -- ISA 10.2 aperture map),
// which is exactly the VDST operand the async-to-LDS instruction wants.
// (Verified in round-1 asm: async VDST and ds_load addresses coincide.)
__device__ __forceinline__ uint32_t lds_addr_of(const void* p) {
  return (uint32_t)(uintptr_t)p;
}

__global__ __launch_bounds__(TPB) void clusterloss_rowmax_kernel(
    const float* __restrict__ md,      // [B_ROWS, P_COLS]
    const float* __restrict__ labels,  // [B_ROWS]
    const float* __restrict__ pc,      // [P_COLS]
    float* __restrict__ rowvals)       // [B_ROWS] out: 128 - rowmax
{
  __shared__ __align__(16) float buf[2][CHUNK];   // double buffer, 2 x 4 KB
  __shared__ float wred[TPB / 32];

  const int   row = blockIdx.x;
  const int   t   = threadIdx.x;
  const float lab = labels[row];
  const float* rowp = md + (size_t)row * P_COLS;

  const uint32_t lds0 = lds_addr_of(&buf[0][t * VEC]);
  const uint32_t lds1 = lds_addr_of(&buf[1][t * VEC]);

  // Prologue: async-stage chunk 0 into LDS buffer 0 (1 instr per wave,
  // 16 B per lane, 512 B per wave, fully coalesced).
  {
    uint64_t g = (uint64_t)(uintptr_t)(rowp + t * VEC);
    asm volatile("global_load_async_to_lds_b128 %0, %1, off"
                 :: "v"(lds0), "v"(g) : "memory");
  }

  float m = 0.0f;   // all masked values are >= 0, so 0 is a valid identity

#pragma unroll
  for (int c = 0; c < NCHUNK; ++c) {
    if (c + 1 < NCHUNK) {
      // WAR guard: DS reads of the other buffer (iteration c-1) must have
      // retired before the async engine may overwrite it (DS and async LDS
      // traffic are mutually unordered per ISA async-memory rules).
      asm volatile("s_wait_dscnt 0x0" ::: "memory");
      const uint32_t lds_nxt = ((c + 1) & 1) ? lds1 : lds0;
      uint64_t g = (uint64_t)(uintptr_t)(rowp + (c + 1) * CHUNK + t * VEC);
      asm volatile("global_load_async_to_lds_b128 %0, %1, off"
                   :: "v"(lds_nxt), "v"(g) : "memory");
      // two async ops in flight; <=1 outstanding  ==>  chunk c has landed
      asm volatile("s_wait_asynccnt 0x1" ::: "memory");
    } else {
      asm volatile("s_wait_asynccnt 0x0" ::: "memory");
    }

    // Each lane consumes exactly the 16 B it staged -> wave-synchronous,
    // no cross-wave barrier required.
    const float4 vv = *(const float4*)&buf[c & 1][t * VEC];
    const float4 qq = *(const float4*)&pc[c * CHUNK + t * VEC];

    const float a0 = (__builtin_fabsf(qq.x - lab) <= DELTA) ? (MAX_DIST - vv.x) : 0.0f;
    const float a1 = (__builtin_fabsf(qq.y - lab) <= DELTA) ? (MAX_DIST - vv.y) : 0.0f;
    const float a2 = (__builtin_fabsf(qq.z - lab) <= DELTA) ? (MAX_DIST - vv.z) : 0.0f;
    const float a3 = (__builtin_fabsf(qq.w - lab) <= DELTA) ? (MAX_DIST - vv.w) : 0.0f;
    m = fmaxf(m, fmaxf(fmaxf(a0, a1), fmaxf(a2, a3)));
  }

  // wave32 max reduction
#pragma unroll
  for (int off = 16; off >= 1; off >>= 1)
    m = fmaxf(m, __shfl_xor(m, off, 32));
  if ((t & 31) == 0) wred[t >> 5] = m;
  __syncthreads();
  if (t == 0) {
    float r = wred[0];
#pragma unroll
    for (int w = 1; w < TPB / 32; ++w) r = fmaxf(r, wred[w]);
    rowvals[row] = MAX_DIST - r;
  }
}

__global__ __launch_bounds__(TPB) void clusterloss_mean_kernel(
    const float* __restrict__ rowvals, float* __restrict__ out)
{
  __shared__ float wsum[TPB / 32];
  const int t = threadIdx.x;
  float s = 0.0f;
  for (int i = t; i < B_ROWS; i += TPB) s += rowvals[i];   // fixed order: deterministic
#pragma unroll
  for (int off = 16; off >= 1; off >>= 1)
    s += __shfl_xor(s, off, 32);
  if ((t & 31) == 0) wsum[t >> 5] = s;
  __syncthreads();
  if (t == 0) {
    float tot = 0.0f;
#pragma unroll
    for (int w = 0; w < TPB / 32; ++w) tot += wsum[w];
    out[0] = tot * (1.0f / (float)B_ROWS);
  }
}

extern "C" void kernel_launch(void* const* d_in, const int* in_sizes, int n_in,
                              void* d_out, int out_size, void* d_ws, size_t ws_size,
                              hipStream_t stream) {
  (void)in_sizes; (void)n_in; (void)out_size; (void)ws_size;
  const float* md     = (const float*)d_in[0];   // [8192*4096] f32
  const float* labels = (const float*)d_in[1];   // [8192]      f32
  const float* pc     = (const float*)d_in[2];   // [4096]      f32
  float* rowvals = (float*)d_ws;                 // 8192 f32 scratch
  float* out     = (float*)d_out;                // 1 f32

  clusterloss_rowmax_kernel<<<dim3(B_ROWS), dim3(TPB), 0, stream>>>(md, labels, pc, rowvals);
  clusterloss_mean_kernel<<<dim3(1), dim3(TPB), 0, stream>>>(rowvals, out);
}